// FRGCN_9096740733056
// MI455X (gfx1250) — compile-verified
//
#include <hip/hip_runtime.h>
#include <hip/hip_bf16.h>

typedef __attribute__((ext_vector_type(2))) float v2f;
typedef __attribute__((ext_vector_type(8))) float v8f;

// Problem constants (match reference)
#define NN   204800
#define EE   3276800
#define BB   4096
#define NPG  50          // nodes per graph (N/B); graph_id = arange//50 -> contiguous
#define RC   3
#define NBC  2

// ---------------------------------------------------------------------------
// WMMA: D = A(16x4 f32) * B(4x16 f32) + C(16x16 f32), wave32.
// Layouts per CDNA5 ISA 7.12.2:
//   A: (lane&15)=M, (lane>>4) selects K pair; VGPR0=Klo, VGPR1=Khi
//   B: (lane&15)=N, (lane>>4) selects K pair
//   C/D: (lane&15)=N, VGPR v -> M = v + 8*(lane>>4)
// ---------------------------------------------------------------------------
static __device__ inline v8f wmma_f32_16x16x4(v2f a, v2f b, v8f c) {
#if __has_builtin(__builtin_amdgcn_wmma_f32_16x16x4_f32)
  return __builtin_amdgcn_wmma_f32_16x16x4_f32(false, a, false, b, (short)0, c,
                                               false, false);
#else
  c[0] += a.x * b.x + a.y * b.y;  // unreachable on gfx1250 per probe
  return c;
#endif
}

// ---------------------------------------------------------------------------
// Fused RGCN node transform: one A read feeds 4 weight matrices
// (3 relation weights + self weight), each [K,32], packed in Wall[4][K][32].
//   xr[n][r][:]  = h[n] @ W_r   (r = 0..2), row stride 96
//   selfb[n][:]  = h[n] @ Wself,            row stride 32
// ---------------------------------------------------------------------------
template <int K, int LDA>
__global__ __launch_bounds__(256) void rgcn_transform(
    const float* __restrict__ A, const float* __restrict__ Wall,
    float* __restrict__ xr, float* __restrict__ selfb, int mtiles) {
  int wave = blockIdx.x * (blockDim.x >> 5) + (threadIdx.x >> 5);
  if (wave >= mtiles) return;  // wave-uniform: EXEC all-ones for WMMA
  const int lane = threadIdx.x & 31;
  const int mn = lane & 15;
  const int khi = lane >> 4;
  const float* arow = A + (size_t)(wave * 16 + mn) * LDA + khi * 2;
  v8f acc[4][2] = {};
  for (int k = 0; k < K; k += 4) {
    v2f a = *(const v2f*)(arow + k);  // 8B-aligned: LDA even, k%4==0, khi*2 even
    const float* wb = Wall + (size_t)(k + khi * 2) * 32 + mn;
#pragma unroll
    for (int m = 0; m < 4; ++m) {
#pragma unroll
      for (int t = 0; t < 2; ++t) {
        v2f b;
        b.x = wb[m * K * 32 + t * 16];
        b.y = wb[m * K * 32 + 32 + t * 16];
        acc[m][t] = wmma_f32_16x16x4(a, b, acc[m][t]);
      }
    }
  }
  const int rbase = wave * 16 + khi * 8;
#pragma unroll
  for (int m = 0; m < 3; ++m)
#pragma unroll
    for (int t = 0; t < 2; ++t)
#pragma unroll
      for (int v = 0; v < 8; ++v)
        xr[(size_t)(rbase + v) * 96 + m * 32 + t * 16 + mn] = acc[m][t][v];
#pragma unroll
  for (int t = 0; t < 2; ++t)
#pragma unroll
    for (int v = 0; v < 8; ++v)
      selfb[(size_t)(rbase + v) * 32 + t * 16 + mn] = acc[3][t][v];
}

// Generic GEMM: C[M,NOUT] = A[M,K] @ W[K,NOUT], compile-time strides.
template <int K, int NOUT, int LDA, int LDW, int LDC>
__global__ __launch_bounds__(256) void gemm16_wmma(const float* __restrict__ A,
                                                   const float* __restrict__ W,
                                                   float* __restrict__ C,
                                                   int mtiles) {
  int wave = blockIdx.x * (blockDim.x >> 5) + (threadIdx.x >> 5);
  if (wave >= mtiles) return;
  const int lane = threadIdx.x & 31;
  const int mn = lane & 15;
  const int khi = lane >> 4;
  const float* arow = A + (size_t)(wave * 16 + mn) * LDA + khi * 2;
  constexpr int NT = NOUT / 16;
  v8f acc[NT] = {};
  for (int k = 0; k < K; k += 4) {
    v2f a = *(const v2f*)(arow + k);
    const float* wrow = W + (size_t)(k + khi * 2) * LDW + mn;
#pragma unroll
    for (int t = 0; t < NT; ++t) {
      v2f b;
      b.x = wrow[t * 16];
      b.y = wrow[LDW + t * 16];
      acc[t] = wmma_f32_16x16x4(a, b, acc[t]);
    }
  }
  float* crow = C + (size_t)(wave * 16 + khi * 8) * LDC + mn;
#pragma unroll
  for (int t = 0; t < NT; ++t)
#pragma unroll
    for (int v = 0; v < 8; ++v) crow[(size_t)v * LDC + t * 16] = acc[t][v];
}

// Dual-weight GEMM (shared A operand): C0 = A@W0, C1 = A@W1.
template <int K, int NOUT, int LDA, int LDW, int LDC>
__global__ __launch_bounds__(256) void gemm16_wmma_dual(
    const float* __restrict__ A, const float* __restrict__ W0,
    const float* __restrict__ W1, float* __restrict__ C0,
    float* __restrict__ C1, int mtiles) {
  int wave = blockIdx.x * (blockDim.x >> 5) + (threadIdx.x >> 5);
  if (wave >= mtiles) return;
  const int lane = threadIdx.x & 31;
  const int mn = lane & 15;
  const int khi = lane >> 4;
  const float* arow = A + (size_t)(wave * 16 + mn) * LDA + khi * 2;
  constexpr int NT = NOUT / 16;
  v8f acc0[NT] = {}, acc1[NT] = {};
  for (int k = 0; k < K; k += 4) {
    v2f a = *(const v2f*)(arow + k);
    const float* w0 = W0 + (size_t)(k + khi * 2) * LDW + mn;
    const float* w1 = W1 + (size_t)(k + khi * 2) * LDW + mn;
#pragma unroll
    for (int t = 0; t < NT; ++t) {
      v2f b0, b1;
      b0.x = w0[t * 16];
      b0.y = w0[LDW + t * 16];
      b1.x = w1[t * 16];
      b1.y = w1[LDW + t * 16];
      acc0[t] = wmma_f32_16x16x4(a, b0, acc0[t]);
      acc1[t] = wmma_f32_16x16x4(a, b1, acc1[t]);
    }
  }
  float* c0 = C0 + (size_t)(wave * 16 + khi * 8) * LDC + mn;
  float* c1 = C1 + (size_t)(wave * 16 + khi * 8) * LDC + mn;
#pragma unroll
  for (int t = 0; t < NT; ++t)
#pragma unroll
    for (int v = 0; v < 8; ++v) {
      c0[(size_t)v * LDC + t * 16] = acc0[t][v];
      c1[(size_t)v * LDC + t * 16] = acc1[t][v];
    }
}

// Wall[m][i][o]: m<3 -> sum_b coef[m,b]*bases[b,i,o] ; m==3 -> wself[i,o]
__global__ void pack_w_all(const float* __restrict__ bases,
                           const float* __restrict__ coef,
                           const float* __restrict__ wself,
                           float* __restrict__ Wall, int IN) {
  int idx = blockIdx.x * blockDim.x + threadIdx.x;
  if (idx >= 4 * IN * 32) return;
  int m = idx / (IN * 32);
  int io = idx - m * (IN * 32);
  float s;
  if (m < 3) {
    s = 0.f;
#pragma unroll
    for (int b = 0; b < NBC; ++b)
      s += coef[m * NBC + b] * bases[b * IN * 32 + io];
  } else {
    s = wself[io];
  }
  Wall[idx] = s;
}

__global__ void zero_i(int* __restrict__ p, int n) {
  int i = blockIdx.x * blockDim.x + threadIdx.x;
  if (i < n) p[i] = 0;
}
__global__ void set_zero_scalar(float* p) { *p = 0.f; }

// ---------------- dst-CSR build (once per launch; edge list is static) -----
__global__ void csr_hist(const int* __restrict__ dst, int* __restrict__ counts) {
  int e = blockIdx.x * blockDim.x + threadIdx.x;
  if (e < EE) atomicAdd(&counts[dst[e]], 1);
}

__global__ void scan_blocks(const int* __restrict__ counts,
                            int* __restrict__ offs, int* __restrict__ bsum) {
  __shared__ int s[256];
  int i = blockIdx.x * 256 + threadIdx.x;
  int v = counts[i];
  s[threadIdx.x] = v;
  __syncthreads();
  for (int d = 1; d < 256; d <<= 1) {  // Hillis-Steele inclusive scan
    int t = (threadIdx.x >= d) ? s[threadIdx.x - d] : 0;
    __syncthreads();
    s[threadIdx.x] += t;
    __syncthreads();
  }
  offs[i] = s[threadIdx.x] - v;  // block-local exclusive
  if (threadIdx.x == 255) bsum[blockIdx.x] = s[255];
}

__global__ void scan_carry(int* __restrict__ bsum, int n) {
  if (blockIdx.x == 0 && threadIdx.x == 0) {
    int run = 0;
    for (int i = 0; i < n; ++i) {
      int t = bsum[i];
      bsum[i] = run;
      run += t;
    }
  }
}

__global__ void add_off(int* __restrict__ offs, const int* __restrict__ bsum) {
  int i = blockIdx.x * 256 + threadIdx.x;
  offs[i] += bsum[blockIdx.x];
}

__global__ void csr_fill(const int* __restrict__ dst,
                         const int* __restrict__ offs, int* __restrict__ cursor,
                         int* __restrict__ ei) {
  int e = blockIdx.x * blockDim.x + threadIdx.x;
  if (e >= EE) return;
  int d = dst[e];
  int pos = offs[d] + atomicAdd(&cursor[d], 1);
  ei[pos] = e;
}

// Canonicalize bucket order (sort by edge id) -> bitwise-deterministic sums;
// then pack (src<<2|etype) and edge_norm in CSR order. One-time, deg~16.
__global__ void csr_sortpack(const int* __restrict__ offs,
                             const int* __restrict__ counts,
                             int* __restrict__ ei, const int* __restrict__ src,
                             const int* __restrict__ ety,
                             const float* __restrict__ enorm,
                             int* __restrict__ es, float* __restrict__ ew) {
  int n = blockIdx.x * blockDim.x + threadIdx.x;
  if (n >= NN) return;
  int st = offs[n], deg = counts[n];
  for (int a = 1; a < deg; ++a) {  // insertion sort ascending
    int key = ei[st + a];
    int b = a - 1;
    while (b >= 0 && ei[st + b] > key) {
      ei[st + b + 1] = ei[st + b];
      --b;
    }
    ei[st + b + 1] = key;
  }
  for (int k = 0; k < deg; ++k) {
    int e = ei[st + k];
    es[st + k] = (src[e] << 2) | ety[e];
    ew[st + k] = enorm[e];
  }
}

// ---------------- atomic-free aggregation, fused with tanh + hcat write ----
// One wave per dst node; lane = channel. agg = sum_e xr[src_e, ety_e, c]*w_e.
__global__ __launch_bounds__(256) void agg_finalize(
    const float* __restrict__ xr, const int* __restrict__ es,
    const float* __restrict__ ew, const int* __restrict__ offs,
    const int* __restrict__ counts, const float* __restrict__ selfb,
    const float* __restrict__ bias, float* __restrict__ hcat_col) {
  int gt = blockIdx.x * blockDim.x + threadIdx.x;
  int n = gt >> 5;
  if (n >= NN) return;
  int c = gt & 31;
  int st = offs[n], deg = counts[n];
  float acc = 0.f;
  for (int k = 0; k < deg; ++k) {
    int p = es[st + k];       // wave-uniform
    float w = ew[st + k];     // wave-uniform
    acc += xr[(size_t)(p >> 2) * 96 + (p & 3) * 32 + c] * w;  // coalesced 128B
  }
  float v = tanhf(acc + selfb[(size_t)n * 32 + c] + bias[c]);
  hcat_col[(size_t)n * 128 + c] = v;
}

// Per-graph masked means (triplet anchors == attention tmean)
__global__ __launch_bounds__(128) void segmean(const float* __restrict__ hcat,
                                               const int* __restrict__ ntype,
                                               float* __restrict__ au,
                                               float* __restrict__ ai) {
  int b = blockIdx.x, c = threadIdx.x;
  int base = b * NPG;
  float su = 0.f, si = 0.f, cu = 0.f, ci = 0.f;
  for (int j = 0; j < NPG; ++j) {
    float v = hcat[(size_t)(base + j) * 128 + c];
    if (ntype[base + j] == 0) {
      su += v;
      cu += 1.f;
    } else {
      si += v;
      ci += 1.f;
    }
  }
  au[(size_t)b * 128 + c] = su / (cu + 1e-8f);
  ai[(size_t)b * 128 + c] = si / (ci + 1e-8f);
}

__global__ __launch_bounds__(128) void triplet_loss(
    const float* __restrict__ au, const float* __restrict__ ai,
    float* __restrict__ loss) {
  __shared__ float rp[128], rn[128];
  int b = blockIdx.x, c = threadIdx.x;
  int bp = (b + BB - 1) & (BB - 1);  // an = roll(ai, 1)
  float u = au[(size_t)b * 128 + c];
  float dp = u - ai[(size_t)b * 128 + c];
  float dn = u - ai[(size_t)bp * 128 + c];
  rp[c] = dp * dp;
  rn[c] = dn * dn;
  __syncthreads();
  if (c == 0) {
    float sp = 0.f, sn = 0.f;
    for (int k = 0; k < 128; ++k) {
      sp += rp[k];
      sn += rn[k];
    }
    unsafeAtomicAdd(loss, fmaxf(sp - sn + 1.0f, 0.f) * (1.0f / BB));
  }
}

// Attention softmax pooling over one graph's 50 nodes, 64 dims.
__global__ __launch_bounds__(64) void attn_pool(const float* __restrict__ p,
                                                const float* __restrict__ t,
                                                const int* __restrict__ ntype,
                                                int ttype,
                                                float* __restrict__ xout) {
  __shared__ float ts[64];
  __shared__ float ps[NPG * 65];  // +1 pad to spread banks
  __shared__ float sc[NPG];
  __shared__ float fm[NPG];
  __shared__ float mz[2];
  int b = blockIdx.x, d = threadIdx.x;
  ts[d] = t[(size_t)b * 64 + d];
  for (int j = 0; j < NPG; ++j)
    ps[j * 65 + d] = p[((size_t)b * NPG + j) * 64 + d];
  __syncthreads();
  if (d < NPG) {
    bool m = (ntype[b * NPG + d] == ttype);
    float s = 0.f;
    for (int k = 0; k < 64; ++k) s += ps[d * 65 + k] * ts[k];
    sc[d] = m ? s * 0.125f : -1e9f;  // /sqrt(64)
    fm[d] = m ? 1.f : 0.f;
  }
  __syncthreads();
  if (d == 0) {
    float mm = sc[0];
    for (int j = 1; j < NPG; ++j) mm = fmaxf(mm, sc[j]);
    mz[0] = mm;
  }
  __syncthreads();
  if (d < NPG) sc[d] = expf(sc[d] - mz[0]) * fm[d];
  __syncthreads();
  if (d == 0) {
    float z = 0.f;
    for (int j = 0; j < NPG; ++j) z += sc[j];
    mz[1] = 1.f / (z + 1e-8f);
  }
  __syncthreads();
  float acc = 0.f;
  for (int j = 0; j < NPG; ++j) acc += sc[j] * ps[j * 65 + d];
  xout[(size_t)b * 64 + d] = acc * mz[1];
}

// MLP head: z=[x_u|x_i] -> relu(z@w1+b1) -> sigmoid(@w2+b2)
__global__ __launch_bounds__(64) void mlp_head(
    const float* __restrict__ xu, const float* __restrict__ xi,
    const float* __restrict__ w1, const float* __restrict__ b1,
    const float* __restrict__ w2, const float* __restrict__ b2,
    float* __restrict__ out) {
  __shared__ float z[128];
  __shared__ float zr[64];
  int b = blockIdx.x, d = threadIdx.x;
  z[d] = xu[(size_t)b * 64 + d];
  z[64 + d] = xi[(size_t)b * 64 + d];
  __syncthreads();
  float s = b1[d];
  for (int k = 0; k < 128; ++k) s += z[k] * w1[k * 64 + d];
  zr[d] = fmaxf(s, 0.f) * w2[d];
  __syncthreads();
  if (d == 0) {
    float acc = b2[0];
    for (int k = 0; k < 64; ++k) acc += zr[k];
    out[b] = 1.f / (1.f + expf(-acc));
  }
}

extern "C" void kernel_launch(void* const* d_in, const int* in_sizes, int n_in,
                              void* d_out, int out_size, void* d_ws,
                              size_t ws_size, hipStream_t stream) {
  (void)in_sizes; (void)n_in; (void)out_size; (void)ws_size;
  const float* x = (const float*)d_in[0];
  const float* enorm = (const float*)d_in[1];
  const int* src = (const int*)d_in[2];
  const int* dst = (const int*)d_in[3];
  const int* ety = (const int*)d_in[4];
  // d_in[5] graph_id: implied by node index (arange // 50)
  const int* ntype = (const int*)d_in[6];
  const float* bases0 = (const float*)d_in[7];
  const float* coef0 = (const float*)d_in[8];
  const float* wself0 = (const float*)d_in[9];
  const float* bias0 = (const float*)d_in[10];
  const float* bases_r = (const float*)d_in[11];
  const float* coef_r = (const float*)d_in[12];
  const float* wself_r = (const float*)d_in[13];
  const float* bias_r = (const float*)d_in[14];
  const float* wp_u = (const float*)d_in[15];
  const float* wt_u = (const float*)d_in[16];
  const float* wp_i = (const float*)d_in[17];
  const float* wt_i = (const float*)d_in[18];
  const float* w1 = (const float*)d_in[19];
  const float* b1 = (const float*)d_in[20];
  const float* w2 = (const float*)d_in[21];
  const float* b2 = (const float*)d_in[22];
  float* out = (float*)d_out;  // [0..B-1] = sigmoid outputs, [B] = triplet loss

  // workspace layout
  float* ws = (float*)d_ws;
  float* hcat = ws;                         // N*128
  float* xr = hcat + (size_t)NN * 128;      // N*96 ([N,R,32])
  float* selfb = xr + (size_t)NN * 96;      // N*32
  float* Wall = selfb + (size_t)NN * 32;    // 4*64*32
  float* au = Wall + 8192;                  // B*128
  float* ai = au + (size_t)BB * 128;        // B*128
  float* tu = ai + (size_t)BB * 128;        // B*64
  float* ti = tu + (size_t)BB * 64;         // B*64
  float* xu = ti + (size_t)BB * 64;         // B*64
  float* xi = xu + (size_t)BB * 64;         // B*64
  float* ew = xi + (size_t)BB * 64;         // E floats (CSR edge weights)
  int* iw = (int*)(ew + (size_t)EE);
  int* counts = iw;                         // N
  int* cursor = counts + NN;                // N
  int* offs = cursor + NN;                  // N
  int* bsum = offs + NN;                    // 800 (pad 1024)
  int* ei = bsum + 1024;                    // E (temp: edge ids)
  int* es = ei + EE;                        // E (packed src<<2|ety)
  float* pu = xr;                           // reuse after GNN layers: N*64
  float* pi = xr + (size_t)NN * 64;         // N*64 (extends into selfb, free)

  const dim3 blk(256);
  const int mt_node = NN / 16;  // 12800 row tiles
  const int gb_node = mt_node / 8;
  const int mt_b = BB / 16;
  const int gb_b = mt_b / 8;

  set_zero_scalar<<<1, 1, 0, stream>>>(out + BB);

  // ---- build dst-CSR once (edge structure is layer-invariant) ----
  zero_i<<<NN / 256, blk, 0, stream>>>(counts, NN);
  zero_i<<<NN / 256, blk, 0, stream>>>(cursor, NN);
  csr_hist<<<EE / 256, blk, 0, stream>>>(dst, counts);
  scan_blocks<<<NN / 256, blk, 0, stream>>>(counts, offs, bsum);
  scan_carry<<<1, 1, 0, stream>>>(bsum, NN / 256);
  add_off<<<NN / 256, blk, 0, stream>>>(offs, bsum);
  csr_fill<<<EE / 256, blk, 0, stream>>>(dst, offs, cursor, ei);
  csr_sortpack<<<NN / 256, blk, 0, stream>>>(offs, counts, ei, src, ety, enorm,
                                             es, ew);

  // ---- layer 0 (in=64 -> 32) ----
  pack_w_all<<<(4 * 64 * 32) / 256, blk, 0, stream>>>(bases0, coef0, wself0,
                                                      Wall, 64);
  rgcn_transform<64, 64><<<gb_node, blk, 0, stream>>>(x, Wall, xr, selfb,
                                                      mt_node);
  agg_finalize<<<(NN * 32) / 256, blk, 0, stream>>>(xr, es, ew, offs, counts,
                                                    selfb, bias0, hcat);

  // ---- layers 1..3 (32 -> 32) ----
  for (int l = 1; l < 4; ++l) {
    pack_w_all<<<(4 * 32 * 32) / 256, blk, 0, stream>>>(
        bases_r + (size_t)(l - 1) * NBC * 32 * 32, coef_r + (l - 1) * RC * NBC,
        wself_r + (size_t)(l - 1) * 32 * 32, Wall, 32);
    rgcn_transform<32, 128><<<gb_node, blk, 0, stream>>>(
        hcat + (l - 1) * 32, Wall, xr, selfb, mt_node);
    agg_finalize<<<(NN * 32) / 256, blk, 0, stream>>>(
        xr, es, ew, offs, counts, selfb, bias_r + (l - 1) * 32, hcat + l * 32);
  }

  // ---- per-graph means + triplet loss ----
  segmean<<<BB, 128, 0, stream>>>(hcat, ntype, au, ai);
  triplet_loss<<<BB, 128, 0, stream>>>(au, ai, out + BB);

  // ---- attention projections (WMMA) ----
  gemm16_wmma_dual<128, 64, 128, 64, 64><<<gb_node, blk, 0, stream>>>(
      hcat, wp_u, wp_i, pu, pi, mt_node);
  gemm16_wmma<128, 64, 128, 64, 64><<<gb_b, blk, 0, stream>>>(au, wt_u, tu,
                                                              mt_b);
  gemm16_wmma<128, 64, 128, 64, 64><<<gb_b, blk, 0, stream>>>(ai, wt_i, ti,
                                                              mt_b);

  attn_pool<<<BB, 64, 0, stream>>>(pu, tu, ntype, 0, xu);
  attn_pool<<<BB, 64, 0, stream>>>(pi, ti, ntype, 1, xi);

  // ---- MLP head ----
  mlp_head<<<BB, 64, 0, stream>>>(xu, xi, w1, b1, w2, b2, out);
}